// RWKV_Tmix_x070_22359599743116
// MI455X (gfx1250) — compile-verified
//
#include <hip/hip_runtime.h>
#include <hip/hip_bf16.h>
#include <math.h>

// RWKV-7 Tmix for MI455X (gfx1250): f32 WMMA GEMMs + register-resident delta-rule scan.

typedef __attribute__((ext_vector_type(2))) float v2f;
typedef __attribute__((ext_vector_type(8))) float v8f;

#define ACT_NONE 0
#define ACT_TANH 1
#define ACT_SIGMOID 2
#define ACT_DECAY 3

__device__ __forceinline__ float sigmoidf_(float z) { return 1.0f / (1.0f + expf(-z)); }

// ---------------------------------------------------------------------------
// Generic tiled GEMM, D[M,N] = act( A[M,K] * B + bias ), f32 WMMA 16x16x4.
//  - If mixv != nullptr, A is built on the fly from x / shift_state:
//      A[row,k] = x[row,k] + (prev(row,k) - x[row,k]) * mixv[k]
//  - transB=1: B is W[N,K] row-major (nn.Linear: out = A @ W.T)
//  - transB=0: B is  [K,N] row-major
// Block: 128 threads (4 wave32). Block tile 64(M) x 128(N), KC = 64.
// Each wave owns a 32x64 subtile = 2x4 WMMA accumulators.
// LDS: A row-major [64][66]; B stored n-major [128][66] so both A and B
// fragments are contiguous 8-byte (b64) LDS loads.
// ---------------------------------------------------------------------------
__global__ __launch_bounds__(128) void gemm_wmma_f32(
    const float* __restrict__ A,
    const float* __restrict__ X, const float* __restrict__ SS,
    const float* __restrict__ mixv,
    const float* __restrict__ Bm, int transB, int ldb,
    const float* __restrict__ bias, int act,
    float* __restrict__ D, int M, int N, int K, int T)
{
    __shared__ float As[64 * 66];     // [row][k]  (+2 pad)
    __shared__ float Bs[128 * 66];    // [col][k]  (+2 pad)  Bs[n*66+k] = B[k][n]

    const int tid   = threadIdx.x;
    const int lane  = tid & 31;
    const int wid   = tid >> 5;
    const int waveM = wid & 1;
    const int waveN = wid >> 1;
    const int rowBase = blockIdx.y * 64;
    const int colBase = blockIdx.x * 128;

    v8f acc[2][4] = {};

    for (int k0 = 0; k0 < K; k0 += 64) {
        // --- stage A tile (optionally with fused token-shift mix) ---
        for (int idx = tid; idx < 64 * 64; idx += 128) {
            int rr = idx >> 6, cc = idx & 63;        // cc = k (coalesced)
            int grow = rowBase + rr;
            int gk = k0 + cc;
            float val = 0.0f;
            if (gk < K) {
                if (mixv) {
                    int b = grow / T;
                    int t = grow - b * T;
                    float cur  = X[(size_t)grow * K + gk];
                    float prev = (t > 0) ? X[(size_t)(grow - 1) * K + gk]
                                         : SS[(size_t)b * K + gk];
                    val = cur + (prev - cur) * mixv[gk];
                } else {
                    val = A[(size_t)grow * K + gk];
                }
            }
            As[rr * 66 + cc] = val;
        }
        // --- stage B tile as Bs[n][k] (coalesced global reads both layouts) ---
        if (transB) {
            for (int idx = tid; idx < 128 * 64; idx += 128) {
                int nn = idx >> 6, kk2 = idx & 63;   // kk2 inner: W row is contiguous in k
                int gk = k0 + kk2;
                int gc = colBase + nn;
                float val = 0.0f;
                if (gk < K && gc < N)
                    val = Bm[(size_t)gc * ldb + gk];
                Bs[nn * 66 + kk2] = val;
            }
        } else {
            for (int idx = tid; idx < 128 * 64; idx += 128) {
                int kk2 = idx >> 6, nn = idx & 127;  // nn inner: B row contiguous in n
                int gk = k0 + kk2;
                int gc = colBase + nn;
                float val = 0.0f;
                if (gk < K && gc < N)
                    val = Bm[(size_t)gk * ldb + gc];
                Bs[nn * 66 + kk2] = val;
            }
        }
        __syncthreads();

        // --- WMMA inner loop over KC in steps of 4 ---
        const int mrow0 = waveM * 32 + (lane & 15);
        const int ncol0 = waveN * 64 + (lane & 15);
        const int kb0   = (lane >> 4) << 1;   // lanes 0-15 -> K 0,1 ; 16-31 -> K 2,3
        #pragma unroll
        for (int kc = 0; kc < 64; kc += 4) {
            int kb = kc + kb0;
            v2f af[2], bf[4];
            #pragma unroll
            for (int mi = 0; mi < 2; mi++) {
                int m = mrow0 + mi * 16;
                af[mi].x = As[m * 66 + kb];
                af[mi].y = As[m * 66 + kb + 1];
            }
            #pragma unroll
            for (int ni = 0; ni < 4; ni++) {
                int n = ncol0 + ni * 16;
                bf[ni].x = Bs[n * 66 + kb];
                bf[ni].y = Bs[n * 66 + kb + 1];
            }
            #pragma unroll
            for (int mi = 0; mi < 2; mi++)
                #pragma unroll
                for (int ni = 0; ni < 4; ni++)
                    acc[mi][ni] = __builtin_amdgcn_wmma_f32_16x16x4_f32(
                        false, af[mi], false, bf[ni], (short)0, acc[mi][ni],
                        false, false);
        }
        __syncthreads();
    }

    // --- epilogue: C/D layout -> lane (n = lane&15, m = i + 8*(lane>>4)) ---
    #pragma unroll
    for (int mi = 0; mi < 2; mi++) {
        #pragma unroll
        for (int ni = 0; ni < 4; ni++) {
            #pragma unroll
            for (int i = 0; i < 8; i++) {
                int row = rowBase + waveM * 32 + mi * 16 + i + 8 * (lane >> 4);
                int col = colBase + waveN * 64 + ni * 16 + (lane & 15);
                if (col < N) {
                    float val = acc[mi][ni][i];
                    if (act == ACT_TANH) {
                        val = tanhf(val);
                    } else if (act == ACT_SIGMOID) {
                        float z = (bias ? bias[col] : 0.0f) + val;
                        val = sigmoidf_(z);
                    } else if (act == ACT_DECAY) {
                        // w = -softplus(-(w0 + v)) - 0.5, numerically stable
                        float z = bias[col] + val;
                        float t_ = -z;
                        float sp = fmaxf(t_, 0.0f) + log1pf(expf(-fabsf(t_)));
                        val = -sp - 0.5f;
                    }
                    D[(size_t)row * N + col] = val;
                }
            }
        }
    }
}

// ---------------------------------------------------------------------------
// Per-head prep: kk = normalize(k*k_k), k *= (1+(a-1)*k_a), v = lerp via vsig.
// One block of 64 threads per (b,t,h).
// ---------------------------------------------------------------------------
__global__ __launch_bounds__(64) void prep_kernel(
    float* __restrict__ kbuf, const float* __restrict__ abuf,
    const float* __restrict__ vfirst, float* __restrict__ vbuf,
    const float* __restrict__ vsig,
    const float* __restrict__ k_k, const float* __restrict__ k_a,
    float* __restrict__ kkout, int H)
{
    int j = threadIdx.x;
    int hid = blockIdx.x;           // (b*T + t)*H + h
    int h = hid % H;
    size_t base = (size_t)hid * 64;
    int c = h * 64 + j;

    float kr  = kbuf[base + j];
    float kkv = kr * k_k[c];

    __shared__ float red[64];
    red[j] = kkv * kkv;
    __syncthreads();
    for (int s = 32; s > 0; s >>= 1) {
        if (j < s) red[j] += red[j + s];
        __syncthreads();
    }
    float denom = fmaxf(sqrtf(red[0]), 1e-12f);
    kkout[base + j] = kkv / denom;

    float aj = abuf[base + j];
    kbuf[base + j] = kr * (1.0f + (aj - 1.0f) * k_a[c]);

    float vr = vbuf[base + j];
    vbuf[base + j] = vr + (vfirst[base + j] - vr) * vsig[base + j];
}

// ---------------------------------------------------------------------------
// Sequential delta-rule scan. One block of 64 threads per (b,h); thread i
// owns state row S[i,0..63] in registers. Per-step vectors staged in LDS.
//   S = S*diag(exp(w)) + (S a) b^T + v k^T ;  o = S r ;  a=-kk, b=kk*a_in
// ---------------------------------------------------------------------------
__global__ __launch_bounds__(64) void scan_kernel(
    const float* __restrict__ r, const float* __restrict__ w,
    const float* __restrict__ k, const float* __restrict__ v,
    const float* __restrict__ kkb, const float* __restrict__ ab,
    const float* __restrict__ S0, float* __restrict__ o, int T, int H)
{
    int i = threadIdx.x;
    int b = blockIdx.x / H;
    int h = blockIdx.x % H;

    float S[64];
    size_t s0base = ((size_t)(b * H + h) * 64 + i) * 64;
    #pragma unroll
    for (int j = 0; j < 64; j++) S[j] = S0[s0base + j];

    __shared__ float aL[64], bL[64], kL[64], ewL[64], rL[64];

    for (int t = 0; t < T; t++) {
        size_t base = ((size_t)(b * T + t) * H + h) * 64;
        float kkj = kkb[base + i];
        float aj  = ab[base + i];
        aL[i]  = -kkj;
        bL[i]  = kkj * aj;
        kL[i]  = k[base + i];
        ewL[i] = expf(w[base + i]);
        rL[i]  = r[base + i];
        float vi = v[base + i];
        __syncthreads();

        float sa = 0.0f;
        #pragma unroll
        for (int j = 0; j < 64; j++) sa += S[j] * aL[j];

        float oi = 0.0f;
        #pragma unroll
        for (int j = 0; j < 64; j++) {
            float sj = S[j] * ewL[j] + sa * bL[j] + vi * kL[j];
            S[j] = sj;
            oi += sj * rL[j];
        }
        o[base + i] = oi;
        __syncthreads();
    }
}

// ---------------------------------------------------------------------------
// GroupNorm(H groups) + bonus term + gate. One 64-thread block per (b,t,h).
// ---------------------------------------------------------------------------
__global__ __launch_bounds__(64) void post_kernel(
    const float* __restrict__ o, const float* __restrict__ r,
    const float* __restrict__ k, const float* __restrict__ v,
    const float* __restrict__ g, const float* __restrict__ r_k,
    const float* __restrict__ ln_w, const float* __restrict__ ln_b,
    float* __restrict__ o2, int H)
{
    int j = threadIdx.x;
    int hid = blockIdx.x;
    int h = hid % H;
    size_t base = (size_t)hid * 64;
    int c = h * 64 + j;

    float oj  = o[base + j];
    float rkj = r[base + j] * k[base + j] * r_k[c];

    __shared__ float s1[64], s2[64], s3[64];
    s1[j] = oj; s2[j] = oj * oj; s3[j] = rkj;
    __syncthreads();
    for (int s = 32; s > 0; s >>= 1) {
        if (j < s) { s1[j] += s1[j + s]; s2[j] += s2[j + s]; s3[j] += s3[j + s]; }
        __syncthreads();
    }
    float mu  = s1[0] * (1.0f / 64.0f);
    float var = s2[0] * (1.0f / 64.0f) - mu * mu;
    float gn  = (oj - mu) * rsqrtf(var + 6.4e-4f) * ln_w[c] + ln_b[c];  // eps = 1e-5*64
    o2[base + j] = (gn + s3[0] * v[base + j]) * g[base + j];
}

// ---------------------------------------------------------------------------
extern "C" void kernel_launch(void* const* d_in, const int* in_sizes, int n_in,
                              void* d_out, int out_size, void* d_ws, size_t ws_size,
                              hipStream_t stream)
{
    (void)in_sizes; (void)n_in; (void)out_size; (void)ws_size;

    const int Bb = 2, Tt = 2048, Cc = 2048, Hh = 32;
    const int Mr = Bb * Tt;                 // 4096 rows
    const size_t S = (size_t)Mr * Cc;       // 8.39M elements

    const float* x     = (const float*)d_in[0];
    const float* ss    = (const float*)d_in[1];
    const float* vfst  = (const float*)d_in[2];
    const float* wkv0  = (const float*)d_in[3];
    const float* x_r   = (const float*)d_in[4];
    const float* x_w   = (const float*)d_in[5];
    const float* x_k   = (const float*)d_in[6];
    const float* x_v   = (const float*)d_in[7];
    const float* x_a   = (const float*)d_in[8];
    const float* x_g   = (const float*)d_in[9];
    const float* w0    = (const float*)d_in[10];
    const float* w1    = (const float*)d_in[11];
    const float* w2    = (const float*)d_in[12];
    const float* a0    = (const float*)d_in[13];
    const float* a1    = (const float*)d_in[14];
    const float* a2    = (const float*)d_in[15];
    const float* v0    = (const float*)d_in[16];
    const float* v1    = (const float*)d_in[17];
    const float* v2w   = (const float*)d_in[18];
    const float* g1    = (const float*)d_in[19];
    const float* g2    = (const float*)d_in[20];
    const float* k_k   = (const float*)d_in[21];
    const float* k_a   = (const float*)d_in[22];
    const float* r_k   = (const float*)d_in[23];
    const float* Wr    = (const float*)d_in[24];
    const float* Wk    = (const float*)d_in[25];
    const float* Wv    = (const float*)d_in[26];
    const float* Wo    = (const float*)d_in[27];
    const float* ln_w  = (const float*)d_in[28];
    const float* ln_b  = (const float*)d_in[29];

    float* ws = (float*)d_ws;
    float* r_buf  = ws + 0 * S;
    float* w_buf  = ws + 1 * S;
    float* k_buf  = ws + 2 * S;
    float* v_buf  = ws + 3 * S;
    float* vsig   = ws + 4 * S;
    float* a_buf  = ws + 5 * S;
    float* g_buf  = ws + 6 * S;
    float* kk_buf = ws + 7 * S;
    float* o_buf  = ws + 8 * S;
    float* o2_buf = ws + 9 * S;
    float* wh = ws + 10 * S;                    // 4096 x 64
    float* ah = wh + (size_t)Mr * 64;           // 4096 x 64
    float* vh = ah + (size_t)Mr * 64;           // 4096 x 32
    float* gh = vh + (size_t)Mr * 32;           // 4096 x 128

    dim3 blk(128);
    auto grd = [](int M_, int N_) { return dim3((N_ + 127) / 128, (M_ + 63) / 64); };

    // r = xr @ Wr.T
    gemm_wmma_f32<<<grd(Mr, Cc), blk, 0, stream>>>(nullptr, x, ss, x_r, Wr, 1, Cc,
        nullptr, ACT_NONE, r_buf, Mr, Cc, Cc, Tt);
    // wh = tanh(xw @ w1)
    gemm_wmma_f32<<<grd(Mr, 64), blk, 0, stream>>>(nullptr, x, ss, x_w, w1, 0, 64,
        nullptr, ACT_TANH, wh, Mr, 64, Cc, Tt);
    // w = -softplus(-(w0 + wh @ w2)) - 0.5
    gemm_wmma_f32<<<grd(Mr, Cc), blk, 0, stream>>>(wh, nullptr, nullptr, nullptr, w2, 0, Cc,
        w0, ACT_DECAY, w_buf, Mr, Cc, 64, Tt);
    // k(raw) = xk @ Wk.T
    gemm_wmma_f32<<<grd(Mr, Cc), blk, 0, stream>>>(nullptr, x, ss, x_k, Wk, 1, Cc,
        nullptr, ACT_NONE, k_buf, Mr, Cc, Cc, Tt);
    // v(raw) = xv @ Wv.T
    gemm_wmma_f32<<<grd(Mr, Cc), blk, 0, stream>>>(nullptr, x, ss, x_v, Wv, 1, Cc,
        nullptr, ACT_NONE, v_buf, Mr, Cc, Cc, Tt);
    // vh = xv @ v1
    gemm_wmma_f32<<<grd(Mr, 32), blk, 0, stream>>>(nullptr, x, ss, x_v, v1, 0, 32,
        nullptr, ACT_NONE, vh, Mr, 32, Cc, Tt);
    // vsig = sigmoid(v0 + vh @ v2)
    gemm_wmma_f32<<<grd(Mr, Cc), blk, 0, stream>>>(vh, nullptr, nullptr, nullptr, v2w, 0, Cc,
        v0, ACT_SIGMOID, vsig, Mr, Cc, 32, Tt);
    // ah = xa @ a1
    gemm_wmma_f32<<<grd(Mr, 64), blk, 0, stream>>>(nullptr, x, ss, x_a, a1, 0, 64,
        nullptr, ACT_NONE, ah, Mr, 64, Cc, Tt);
    // a = sigmoid(a0 + ah @ a2)
    gemm_wmma_f32<<<grd(Mr, Cc), blk, 0, stream>>>(ah, nullptr, nullptr, nullptr, a2, 0, Cc,
        a0, ACT_SIGMOID, a_buf, Mr, Cc, 64, Tt);
    // gh = sigmoid(xg @ g1)
    gemm_wmma_f32<<<grd(Mr, 128), blk, 0, stream>>>(nullptr, x, ss, x_g, g1, 0, 128,
        nullptr, ACT_SIGMOID, gh, Mr, 128, Cc, Tt);
    // g = gh @ g2
    gemm_wmma_f32<<<grd(Mr, Cc), blk, 0, stream>>>(gh, nullptr, nullptr, nullptr, g2, 0, Cc,
        nullptr, ACT_NONE, g_buf, Mr, Cc, 128, Tt);

    // per-head normalize kk, scale k, blend v
    prep_kernel<<<Mr * Hh, 64, 0, stream>>>(k_buf, a_buf, vfst, v_buf, vsig,
                                            k_k, k_a, kk_buf, Hh);
    // sequential delta-rule recurrence
    scan_kernel<<<Bb * Hh, 64, 0, stream>>>(r_buf, w_buf, k_buf, v_buf, kk_buf,
                                            a_buf, wkv0, o_buf, Tt, Hh);
    // groupnorm + bonus + gate
    post_kernel<<<Mr * Hh, 64, 0, stream>>>(o_buf, r_buf, k_buf, v_buf, g_buf,
                                            r_k, ln_w, ln_b, o2_buf, Hh);
    // out = o2 @ Wo.T
    gemm_wmma_f32<<<grd(Mr, Cc), blk, 0, stream>>>(o2_buf, nullptr, nullptr, nullptr, Wo, 1, Cc,
        nullptr, ACT_NONE, (float*)d_out, Mr, Cc, Cc, Tt);
}